// MultiPauliQCNN_73186242724572
// MI455X (gfx1250) — compile-verified
//
#include <hip/hip_runtime.h>

typedef float v2f __attribute__((ext_vector_type(2)));
typedef float v8f __attribute__((ext_vector_type(8)));

#define NPATCH 196
#define TILES_PER_IMG 13      // ceil(196/16)
#define WAVES_PER_BLOCK 8
#define PI_F 3.14159265358979323846f

// ---------------- fixed-circuit gates (register-resident 16-amp state) ----------------

template <int MASK>
__device__ __forceinline__ void gate_ry(float2 (&a)[16], float w) {
  float s, c;
  __sincosf(0.5f * w, &s, &c);
#pragma unroll
  for (int k = 0; k < 16; ++k) {
    if (k & MASK) continue;
    float2 a0 = a[k], a1 = a[k | MASK];
    a[k]        = make_float2(c * a0.x - s * a1.x, c * a0.y - s * a1.y);
    a[k | MASK] = make_float2(s * a0.x + c * a1.x, s * a0.y + c * a1.y);
  }
}

template <int CM, int TM>
__device__ __forceinline__ void gate_cx(float2 (&a)[16]) {
#pragma unroll
  for (int k = 0; k < 16; ++k) {
    if ((k & CM) && !(k & TM)) {
      float2 t = a[k];
      a[k] = a[k | TM];
      a[k | TM] = t;
    }
  }
}

template <int CM, int TM>
__device__ __forceinline__ void gate_crx(float2 (&a)[16], float w) {
  float s, c;
  __sincosf(0.5f * w, &s, &c);
#pragma unroll
  for (int k = 0; k < 16; ++k) {
    if ((k & CM) && !(k & TM)) {
      float2 a0 = a[k], a1 = a[k | TM];
      // control=1 block: [[c, -i s],[-i s, c]]
      a[k]      = make_float2(c * a0.x + s * a1.y, c * a0.y - s * a1.x);
      a[k | TM] = make_float2(c * a1.x + s * a0.y, c * a1.y - s * a0.x);
    }
  }
}

// ds_swizzle_b32 group-of-32 XOR butterfly: offset = xor_mask<<10 | and_mask(0x1F)
template <int PAT>
__device__ __forceinline__ float swz(float v) {
  return __int_as_float(__builtin_amdgcn_ds_swizzle(__float_as_int(v), PAT));
}

#define BFLY16(p)                      \
  do {                                 \
    p += swz<(1 << 10) | 0x1F>(p);     \
    p += swz<(2 << 10) | 0x1F>(p);     \
    p += swz<(4 << 10) | 0x1F>(p);     \
    p += swz<(8 << 10) | 0x1F>(p);     \
  } while (0)

// qubit q <-> bitmask (8 >> q): q0=8, q1=4, q2=2, q3=1
__global__ void __launch_bounds__(256) build_s_kernel(const float* __restrict__ qw,
                                                      float* __restrict__ S) {
  __shared__ float2 U[16][16];  // U[k][col]
  const int tid = threadIdx.x;
  if (tid < 16) {
    float w[12];
#pragma unroll
    for (int i = 0; i < 12; ++i) w[i] = qw[i];
    float2 a[16];
#pragma unroll
    for (int k = 0; k < 16; ++k) a[k] = make_float2(k == tid ? 1.f : 0.f, 0.f);
    // conv1
    gate_ry<8>(a, w[0]); gate_ry<4>(a, w[1]); gate_cx<8, 4>(a);
    gate_ry<2>(a, w[2]); gate_ry<1>(a, w[3]); gate_cx<2, 1>(a);
    gate_cx<4, 2>(a);    gate_cx<1, 8>(a);
    // pool1
    gate_crx<4, 8>(a, w[4]); gate_crx<1, 2>(a, w[5]);
    // conv2
    gate_ry<8>(a, w[6]); gate_ry<2>(a, w[7]); gate_cx<8, 2>(a);
    gate_ry<8>(a, w[8]); gate_ry<2>(a, w[9]);
    // pool2
    gate_crx<2, 8>(a, w[10]); gate_ry<8>(a, w[11]);
#pragma unroll
    for (int k = 0; k < 16; ++k) U[k][tid] = a[k];
  }
  __syncthreads();
  // S_i[acol][bcol] = Re( sum_k conj(U[k,acol]) * (P_i U)[k,bcol] ), P on qubit0 (bit 8)
  const int acol = tid >> 4, bcol = tid & 15;
  float sx = 0.f, sy = 0.f, sz = 0.f;
#pragma unroll
  for (int k = 0; k < 16; ++k) {
    float2 u  = U[k][acol];
    float2 vx = U[k ^ 8][bcol];
    float2 vz = U[k][bcol];
    float sgn = (k & 8) ? -1.f : 1.f;
    sx += u.x * vx.x + u.y * vx.y;
    sy += sgn * (u.x * vx.y - u.y * vx.x);
    sz += sgn * (u.x * vz.x + u.y * vz.y);
  }
  S[tid] = sx;
  S[256 + tid] = sy;
  S[512 + tid] = sz;
}

__global__ void init_out_kernel(float* __restrict__ out, const float* __restrict__ fcb) {
  int i = blockIdx.x * blockDim.x + threadIdx.x;
  if (i < 512 * 10) out[i] = fcb[i % 10];
}

// ---------------- main kernel: one wave per 16-sample tile ----------------

__global__ void __launch_bounds__(256) qcnn_kernel(const float* __restrict__ x,
                                                   const float* __restrict__ S,
                                                   const float* __restrict__ fcw,
                                                   float* __restrict__ out) {
  __shared__ float2 cs[WAVES_PER_BLOCK][16][4];  // (cos, sin) per sample per qubit

  const int lane  = threadIdx.x & 31;
  const int wv    = threadIdx.x >> 5;
  const int gw    = blockIdx.x * WAVES_PER_BLOCK + wv;  // 0..6655 = 512*13
  const int batch = gw / TILES_PER_IMG;
  const int tile  = gw % TILES_PER_IMG;
  const int half  = lane >> 4;
  const int n     = lane & 15;

  // --- 1) load 2x2 patches, compute (cos, sin) of angle/2 into LDS ---
  {
    const int sl = lane >> 1;  // sample 0..15 within tile
    const int wr = lane & 1;   // row within 2x2 patch
    const int p  = tile * 16 + sl;
    float p0 = 0.f, p1 = 0.f;
    if (p < NPATCH) {
      const int pi = p / 14, pj = p % 14;
      const float* src = x + batch * 784 + (2 * pi + wr) * 28 + 2 * pj;
      p0 = src[0];
      p1 = src[1];
    }
    float c0, s0, c1, s1;
    __sincosf(p0 * (PI_F * 0.5f), &s0, &c0);
    __sincosf(p1 * (PI_F * 0.5f), &s1, &c1);
    cs[wv][sl][wr * 2 + 0] = make_float2(c0, s0);
    cs[wv][sl][wr * 2 + 1] = make_float2(c1, s1);
  }
  __syncthreads();

  // --- 2) B fragments: S_i rows (4x16 f32 B layout: VGPR0 = K {0|2}, VGPR1 = K {1|3}) ---
  v2f B[3][4];
#pragma unroll
  for (int i = 0; i < 3; ++i) {
#pragma unroll
    for (int kc = 0; kc < 4; ++kc) {
      const int r0 = kc * 4 + 2 * half;
      B[i][kc].x = S[i * 256 + r0 * 16 + n];
      B[i][kc].y = S[i * 256 + (r0 + 1) * 16 + n];
    }
  }

  // --- 3) A fragments: row M = n (sample), K pair split across lane halves ---
  float c[4], s[4];
#pragma unroll
  for (int q = 0; q < 4; ++q) {
    float2 t = cs[wv][n][q];
    c[q] = t.x;
    s[q] = t.y;
  }
  v2f A[4];
#pragma unroll
  for (int kc = 0; kc < 4; ++kc) {
    const int d0 = kc * 4 + 2 * half;
    const int d1 = d0 + 1;
    float v0 = ((d0 & 8) ? s[0] : c[0]) * ((d0 & 4) ? s[1] : c[1]) *
               ((d0 & 2) ? s[2] : c[2]) * ((d0 & 1) ? s[3] : c[3]);
    float v1 = ((d1 & 8) ? s[0] : c[0]) * ((d1 & 4) ? s[1] : c[1]) *
               ((d1 & 2) ? s[2] : c[2]) * ((d1 & 1) ? s[3] : c[3]);
    A[kc].x = v0;
    A[kc].y = v1;
  }

  // --- 4) T_i = X * S_i via chained f32 WMMA (K = 16 as 4 x k4 steps) ---
  v8f acc[3];
#pragma unroll
  for (int i = 0; i < 3; ++i) {
    v8f t = {};
#pragma unroll
    for (int kc = 0; kc < 4; ++kc) {
      t = __builtin_amdgcn_wmma_f32_16x16x4_f32(false, A[kc], false, B[i][kc],
                                                (short)0, t, false, false);
    }
    acc[i] = t;
  }

  // --- 5) feat_i[m] = sum_d T_i[m,d] * X[m,d]  (C layout: VGPR j -> M=j+8*half, N=n)
  //        reduce across the 16 lanes of each half via ds_swizzle XOR butterflies ---
  float feat[3][8];
#pragma unroll
  for (int j = 0; j < 8; ++j) {
    const int m = j + 8 * half;
    float cc[4], ss[4];
#pragma unroll
    for (int q = 0; q < 4; ++q) {
      float2 t = cs[wv][m][q];
      cc[q] = t.x;
      ss[q] = t.y;
    }
    float xv = ((n & 8) ? ss[0] : cc[0]) * ((n & 4) ? ss[1] : cc[1]) *
               ((n & 2) ? ss[2] : cc[2]) * ((n & 1) ? ss[3] : cc[3]);
    float p0 = acc[0][j] * xv;
    float p1 = acc[1][j] * xv;
    float p2 = acc[2][j] * xv;
    BFLY16(p0);
    BFLY16(p1);
    BFLY16(p2);
    feat[0][j] = p0;
    feat[1][j] = p1;
    feat[2][j] = p2;
  }

  // --- 6) FC projection + atomic accumulate: lane n (<10) handles class n ---
  if (n < 10) {
#pragma unroll
    for (int j = 0; j < 8; ++j) {
      const int p = tile * 16 + j + 8 * half;
      if (p < NPATCH) {
        const float* w = fcw + n * 588 + p * 3;
        float contrib = feat[0][j] * w[0] + feat[1][j] * w[1] + feat[2][j] * w[2];
        unsafeAtomicAdd(out + batch * 10 + n, contrib);
      }
    }
  }
}

extern "C" void kernel_launch(void* const* d_in, const int* in_sizes, int n_in,
                              void* d_out, int out_size, void* d_ws, size_t ws_size,
                              hipStream_t stream) {
  const float* x   = (const float*)d_in[0];   // (512,1,28,28)
  const float* qw  = (const float*)d_in[1];   // (12,)
  const float* fcw = (const float*)d_in[2];   // (10, 588)
  const float* fcb = (const float*)d_in[3];   // (10,)
  float* out = (float*)d_out;                 // (512, 10)
  float* S = (float*)d_ws;                    // 3 * 256 floats

  build_s_kernel<<<1, 256, 0, stream>>>(qw, S);
  init_out_kernel<<<(5120 + 255) / 256, 256, 0, stream>>>(out, fcb);
  qcnn_kernel<<<832, 256, 0, stream>>>(x, S, fcw, out);  // 832*8 waves = 512*13 tiles
}